// MGGATRecommender_82755429859427
// MI455X (gfx1250) — compile-verified
//
#include <hip/hip_runtime.h>
#include <cmath>

typedef __attribute__((ext_vector_type(2))) float v2f;
typedef __attribute__((ext_vector_type(8))) float v8f;

#define NUSERS  50000
#define NITEMS  50000
#define UIN     256
#define IIN     256
#define LATENT  128
#define FINALD  64
#define NGRAPH  2
#define NEDGE   1000000

// ---------------------------------------------------------------------------
// f32 WMMA 16x16x4 tile helpers (wave32, full EXEC)
//   A layout: lane(0-15)=row M, a[0]=K+0,a[1]=K+1 ; lanes 16-31: a[0]=K+2,a[1]=K+3
//   B layout: half-VGPRs hold the 4 K rows, N striped across lanes
//   C/D     : VGPR r holds M = r (lanes 0-15) / M = r+8 (lanes 16-31)
// ---------------------------------------------------------------------------
__device__ __forceinline__ v8f gemm_tile_f32(const float* __restrict__ A, int lda,
                                             const float* __restrict__ B, int ldb,
                                             int row0, int col0, int K, v8f acc)
{
    const int lane = threadIdx.x & 31;
    const int half = lane >> 4;
    const int r    = lane & 15;
    const float* __restrict__ arow = A + (size_t)(row0 + r) * lda + 2 * half;
    const int bcol = col0 + r;
    for (int kk = 0; kk < K; kk += 4) {
        v2f a = *(const v2f*)(arow + kk);
        const float* bp = B + (size_t)(kk + 2 * half) * ldb + bcol;
        v2f b;
        b[0] = bp[0];
        b[1] = bp[ldb];
        acc = __builtin_amdgcn_wmma_f32_16x16x4_f32(false, a, false, b,
                                                    (short)0, acc, false, false);
    }
    return acc;
}

__device__ __forceinline__ void store_tile(float* __restrict__ C, int ldc,
                                           int row0, int col0, v8f acc)
{
    const int lane = threadIdx.x & 31;
    const int half = lane >> 4;
    const int c    = col0 + (lane & 15);
#pragma unroll
    for (int r8 = 0; r8 < 8; ++r8)
        C[(size_t)(row0 + r8 + 8 * half) * ldc + c] = acc[r8];
}

// ---------------------------------------------------------------------------
// Stage 1: H1 = S @ W1   (N x K  @  K x LATENT), 8 waves = 8 N-tiles / block
// ---------------------------------------------------------------------------
__global__ __launch_bounds__(256)
void k_gemm_h1(const float* __restrict__ S, const float* __restrict__ W,
               float* __restrict__ H1, int K)
{
    const int wave = threadIdx.x >> 5;      // 0..7 -> N tile
    const int row0 = blockIdx.x * 16;
    v8f acc = {};
    acc = gemm_tile_f32(S, K, W, LATENT, row0, wave * 16, K, acc);
    store_tile(H1, LATENT, row0, wave * 16, acc);
}

// alpha_self / alpha_nb : per-row dot with 128-vec
__global__ __launch_bounds__(256)
void k_alpha(const float* __restrict__ H1, const float* __restrict__ a_self,
             const float* __restrict__ a_nb, float* __restrict__ As,
             float* __restrict__ An, int N)
{
    int i = blockIdx.x * blockDim.x + threadIdx.x;
    if (i >= N) return;
    const float4* h  = (const float4*)(H1 + (size_t)i * LATENT);
    const float4* s4 = (const float4*)a_self;
    const float4* n4 = (const float4*)a_nb;
    float ds = 0.f, dn = 0.f;
#pragma unroll
    for (int k = 0; k < LATENT / 4; ++k) {
        float4 hv = h[k], sv = s4[k], nv = n4[k];
        ds += hv.x * sv.x + hv.y * sv.y + hv.z * sv.z + hv.w * sv.w;
        dn += hv.x * nv.x + hv.y * nv.y + hv.z * nv.z + hv.w * nv.w;
    }
    As[i] = ds;
    An[i] = dn;
}

__global__ __launch_bounds__(256)
void k_fill(float* __restrict__ p, float v, int n)
{
    int i = blockIdx.x * blockDim.x + threadIdx.x;
    if (i < n) p[i] = v;
}

// edge pass 1: score = leaky_relu(As[dst] + An[src]); segment max via int atomics
__global__ __launch_bounds__(256)
void k_edge_score(const int* __restrict__ src, const int* __restrict__ dst,
                  const float* __restrict__ As, const float* __restrict__ An,
                  float* __restrict__ scores, float* __restrict__ m, int E)
{
    int e = blockIdx.x * blockDim.x + threadIdx.x;
    if (e >= E) return;
    int s = src[e], d = dst[e];
    float x  = As[d] + An[s];
    float sc = x > 0.f ? x : 0.2f * x;
    scores[e] = sc;
    if (sc >= 0.f) atomicMax((int*)(m + d), __float_as_int(sc));
    else           atomicMin((unsigned int*)(m + d), __float_as_uint(sc));
}

// edge pass 2: e = exp(score - max[dst]); segment sum
__global__ __launch_bounds__(256)
void k_edge_exp(const int* __restrict__ dst, float* __restrict__ scores,
                const float* __restrict__ m, float* __restrict__ ssum, int E)
{
    int e = blockIdx.x * blockDim.x + threadIdx.x;
    if (e >= E) return;
    int d = dst[e];
    float v = expf(scores[e] - m[d]);
    scores[e] = v;
    unsafeAtomicAdd(&ssum[d], v);
}

// edge pass 3: H2[dst] += omega_g * alpha * H1[src]   (one wave per edge)
__global__ __launch_bounds__(256)
void k_edge_scatter(const int* __restrict__ src, const int* __restrict__ dst,
                    const float* __restrict__ scores, const float* __restrict__ ssum,
                    const float* __restrict__ H1, float* __restrict__ H2,
                    const float* __restrict__ omega, int g, int E)
{
    int wave = threadIdx.x >> 5;
    int lane = threadIdx.x & 31;
    int e = blockIdx.x * (blockDim.x >> 5) + wave;
    if (e >= E) return;
    int s = src[e], d = dst[e];
    float coef = omega[g] * scores[e] / (ssum[d] + 1e-16f);
    float4 h = ((const float4*)(H1 + (size_t)s * LATENT))[lane];
    float* out = H2 + (size_t)d * LATENT + lane * 4;
    unsafeAtomicAdd(out + 0, coef * h.x);
    unsafeAtomicAdd(out + 1, coef * h.y);
    unsafeAtomicAdd(out + 2, coef * h.z);
    unsafeAtomicAdd(out + 3, coef * h.w);
}

// Stage 2 (fused): H3 = elu(H2@W2 + S@Ws + bias)
__global__ __launch_bounds__(256)
void k_gemm_h3(const float* __restrict__ H2, const float* __restrict__ W2,
               const float* __restrict__ S, const float* __restrict__ Ws,
               const float* __restrict__ bias, float* __restrict__ H3, int Kin)
{
    const int wave = threadIdx.x >> 5;
    const int row0 = blockIdx.x * 16;
    const int col0 = wave * 16;
    v8f acc = {};
    acc = gemm_tile_f32(H2, LATENT, W2, LATENT, row0, col0, LATENT, acc);
    acc = gemm_tile_f32(S, Kin, Ws, LATENT, row0, col0, Kin, acc);
    const int lane = threadIdx.x & 31;
    const int half = lane >> 4;
    const int c    = col0 + (lane & 15);
    const float b  = bias[c];
#pragma unroll
    for (int r8 = 0; r8 < 8; ++r8) {
        float x = acc[r8] + b;
        float y = x > 0.f ? x : expm1f(x);
        H3[(size_t)(row0 + r8 + 8 * half) * LATENT + c] = y;
    }
}

// Stage 3: out = elu(H3@W3) + H4   (4 N-tiles -> 4 waves / block)
__global__ __launch_bounds__(128)
void k_gemm_final(const float* __restrict__ H3, const float* __restrict__ W3,
                  const float* __restrict__ H4, float* __restrict__ Uall)
{
    const int wave = threadIdx.x >> 5;      // 0..3
    const int row0 = blockIdx.x * 16;
    const int col0 = wave * 16;
    v8f acc = {};
    acc = gemm_tile_f32(H3, LATENT, W3, FINALD, row0, col0, LATENT, acc);
    const int lane = threadIdx.x & 31;
    const int half = lane >> 4;
    const int c    = col0 + (lane & 15);
#pragma unroll
    for (int r8 = 0; r8 < 8; ++r8) {
        float x = acc[r8];
        float y = x > 0.f ? x : expm1f(x);
        size_t idx = (size_t)(row0 + r8 + 8 * half) * FINALD + c;
        Uall[idx] = y + H4[idx];
    }
}

// final prediction: dot(U[u], B[i]) + biases -> scaled sigmoid
__global__ __launch_bounds__(256)
void k_predict(const int* __restrict__ uid, const int* __restrict__ iid,
               const float* __restrict__ U, const float* __restrict__ B,
               const float* __restrict__ bu, const float* __restrict__ bb,
               const float* __restrict__ bx, float* __restrict__ out, int n)
{
    int i = blockIdx.x * blockDim.x + threadIdx.x;
    if (i >= n) return;
    int u = uid[i], b = iid[i];
    const float4* up = (const float4*)(U + (size_t)u * FINALD);
    const float4* bp = (const float4*)(B + (size_t)b * FINALD);
    float dot = 0.f;
#pragma unroll
    for (int k = 0; k < FINALD / 4; ++k) {
        float4 a = up[k], c = bp[k];
        dot += a.x * c.x + a.y * c.y + a.z * c.z + a.w * c.w;
    }
    float raw = dot + bu[u] + bb[b] + bx[0];
    float sig = 1.f / (1.f + expf(-raw));
    out[i] = 4.f * sig + 1.f;
}

// ---------------------------------------------------------------------------
// host side
// ---------------------------------------------------------------------------
static inline float* bump(char*& p, size_t bytes)
{
    float* r = (float*)p;
    p += (bytes + 255) & ~size_t(255);
    return r;
}

static void run_side(const float* S, const int* edges,
                     const float* W1, const float* a_self, const float* a_nb,
                     const float* omega,
                     const float* W2, const float* Ws2w, const float* Ws2b,
                     const float* W3, const float* H4,
                     float* H1, float* H2, float* Out,
                     float* As, float* An, float* m, float* ssum, float* scores,
                     int N, int Kin, hipStream_t stream)
{
    const int mt = N / 16;                          // 3125 exact
    const int nb = (N + 255) / 256;
    const int eb = (NEDGE + 255) / 256;
    const int sb = (NEDGE + 7) / 8;                 // 8 waves/block

    k_fill<<<(N * LATENT + 255) / 256, 256, 0, stream>>>(H2, 0.f, N * LATENT);
    k_gemm_h1<<<mt, 256, 0, stream>>>(S, W1, H1, Kin);
    k_alpha<<<nb, 256, 0, stream>>>(H1, a_self, a_nb, As, An, N);

    for (int g = 0; g < NGRAPH; ++g) {
        const int* src = edges + ((size_t)g * 2 + 0) * NEDGE;
        const int* dst = edges + ((size_t)g * 2 + 1) * NEDGE;
        k_fill<<<nb, 256, 0, stream>>>(m, -INFINITY, N);
        k_fill<<<nb, 256, 0, stream>>>(ssum, 0.f, N);
        k_edge_score<<<eb, 256, 0, stream>>>(src, dst, As, An, scores, m, NEDGE);
        k_edge_exp<<<eb, 256, 0, stream>>>(dst, scores, m, ssum, NEDGE);
        k_edge_scatter<<<sb, 256, 0, stream>>>(src, dst, scores, ssum, H1, H2,
                                               omega, g, NEDGE);
    }

    // H3 overwrites H1 (H1 no longer needed)
    k_gemm_h3<<<mt, 256, 0, stream>>>(H2, W2, S, Ws2w, Ws2b, H1, Kin);
    k_gemm_final<<<mt, 128, 0, stream>>>(H1, W3, H4, Out);
}

extern "C" void kernel_launch(void* const* d_in, const int* in_sizes, int n_in,
                              void* d_out, int out_size, void* d_ws, size_t ws_size,
                              hipStream_t stream)
{
    const int*   uid     = (const int*)  d_in[0];
    const int*   iid     = (const int*)  d_in[1];
    const float* S_u     = (const float*)d_in[2];
    const float* S_b     = (const float*)d_in[3];
    const int*   edges_u = (const int*)  d_in[4];
    const int*   edges_b = (const int*)  d_in[5];
    const float* W1_u    = (const float*)d_in[6];
    const float* aself_u = (const float*)d_in[7];
    const float* anb_u   = (const float*)d_in[8];
    const float* omega_u = (const float*)d_in[9];
    const float* W1_b    = (const float*)d_in[10];
    const float* aself_b = (const float*)d_in[11];
    const float* anb_b   = (const float*)d_in[12];
    const float* omega_b = (const float*)d_in[13];
    const float* Wu2     = (const float*)d_in[14];
    const float* Wus2_w  = (const float*)d_in[15];
    const float* Wus2_b  = (const float*)d_in[16];
    const float* Wb2     = (const float*)d_in[17];
    const float* Wbs2_w  = (const float*)d_in[18];
    const float* Wbs2_b  = (const float*)d_in[19];
    const float* Wu3     = (const float*)d_in[20];
    const float* Wb3     = (const float*)d_in[21];
    const float* Hu4     = (const float*)d_in[22];
    const float* Hb4     = (const float*)d_in[23];
    const float* bu      = (const float*)d_in[24];
    const float* bb      = (const float*)d_in[25];
    const float* bx      = (const float*)d_in[26];

    char* p = (char*)d_ws;
    float* H1     = bump(p, (size_t)NUSERS * LATENT * 4);   // 25.6 MB (also H3)
    float* H2     = bump(p, (size_t)NUSERS * LATENT * 4);   // 25.6 MB
    float* Uall   = bump(p, (size_t)NUSERS * FINALD * 4);   // 12.8 MB
    float* Ball   = bump(p, (size_t)NITEMS * FINALD * 4);   // 12.8 MB
    float* As     = bump(p, (size_t)NUSERS * 4);
    float* An     = bump(p, (size_t)NUSERS * 4);
    float* m      = bump(p, (size_t)NUSERS * 4);
    float* ssum   = bump(p, (size_t)NUSERS * 4);
    float* scores = bump(p, (size_t)NEDGE * 4);             // 4 MB

    run_side(S_u, edges_u, W1_u, aself_u, anb_u, omega_u,
             Wu2, Wus2_w, Wus2_b, Wu3, Hu4,
             H1, H2, Uall, As, An, m, ssum, scores, NUSERS, UIN, stream);

    run_side(S_b, edges_b, W1_b, aself_b, anb_b, omega_b,
             Wb2, Wbs2_w, Wbs2_b, Wb3, Hb4,
             H1, H2, Ball, As, An, m, ssum, scores, NITEMS, IIN, stream);

    const int n = in_sizes[0];
    k_predict<<<(n + 255) / 256, 256, 0, stream>>>(uid, iid, Uall, Ball,
                                                   bu, bb, bx, (float*)d_out, n);
}